// GMMAttention_37993280701087
// MI455X (gfx1250) — compile-verified
//
#include <hip/hip_runtime.h>
#include <hip/hip_bf16.h>

typedef __attribute__((ext_vector_type(2))) float v2f;
typedef __attribute__((ext_vector_type(8))) float v8f;

#define NB     128   // batch
#define T_LEN  2000
#define D_RNN  1024
#define D_EMB  512
#define D_ATT  128
#define MMIX   8

// ---------------------------------------------------------------------------
// Kernel 1: MLP front-end via V_WMMA_F32_16X16X4_F32 (f32-exact matrix path).
// Grid: 8 blocks (16 batch rows each) x 256 threads (8 waves).
// Wave w owns the 16-wide column tile [16w,16w+16). h1/h2 staged in LDS.
// Fragment layout per ISA 7.12.2 (f32, wave32):
//   A 16x4 : lane L holds A[L&15][2*(L>>4) + {0,1}] in {v0,v1}
//   B 4x16 : lane L holds B[2*(L>>4) + {0,1}][L&15] in {v0,v1}
//   C/D    : vgpr r, lane L -> D[r + 8*(L>>4)][L&15]
// ---------------------------------------------------------------------------
__launch_bounds__(256)
__global__ void mlp_wmma_kernel(const float* __restrict__ X,
                                const float* __restrict__ prev_loc,
                                const float* __restrict__ W1, const float* __restrict__ b1,
                                const float* __restrict__ W2, const float* __restrict__ b2,
                                const float* __restrict__ W3,
                                float* __restrict__ out_loc,
                                float* __restrict__ ws_loc,
                                float* __restrict__ ws_inv_std,
                                float* __restrict__ ws_wm)
{
    __shared__ float h1s[16 * D_ATT];   // 8 KB
    __shared__ float h2s[16 * D_ATT];   // 8 KB
    __shared__ float prs[16 * 24];      // params tile

    const int tid   = threadIdx.x;
    const int wave  = tid >> 5;
    const int lane  = tid & 31;
    const int row   = lane & 15;        // M row (A) / N col (B)
    const int hi    = lane >> 4;        // 0: K=0,1  1: K=2,3
    const int rbase = blockIdx.x * 16;  // batch-row base

    // ---- h1 = relu(X @ W1 + b1) : K = 1024, two independent WMMA chains ----
    {
        const int c = wave * 16 + row;
        v8f acc0 = {}, acc1 = {};
        const float2* Xr = (const float2*)(X + (size_t)(rbase + row) * D_RNN);
        for (int k = 0; k < D_RNN; k += 8) {
            float2 av0 = Xr[(k >> 1) + hi];
            float2 av1 = Xr[(k >> 1) + 2 + hi];
            v2f a0, a1, bb0, bb1;
            a0.x = av0.x; a0.y = av0.y;
            a1.x = av1.x; a1.y = av1.y;
            bb0.x = W1[(size_t)(k + 2 * hi)     * D_ATT + c];
            bb0.y = W1[(size_t)(k + 2 * hi + 1) * D_ATT + c];
            bb1.x = W1[(size_t)(k + 4 + 2 * hi)     * D_ATT + c];
            bb1.y = W1[(size_t)(k + 4 + 2 * hi + 1) * D_ATT + c];
            acc0 = __builtin_amdgcn_wmma_f32_16x16x4_f32(false, a0, false, bb0,
                                                         (short)0, acc0, false, false);
            acc1 = __builtin_amdgcn_wmma_f32_16x16x4_f32(false, a1, false, bb1,
                                                         (short)0, acc1, false, false);
        }
        #pragma unroll
        for (int r = 0; r < 8; ++r) {
            float v = acc0[r] + acc1[r] + b1[c];
            h1s[(r + 8 * hi) * D_ATT + c] = v > 0.f ? v : 0.f;
        }
    }
    __syncthreads();

    // ---- h2 = relu(h1 @ W2 + b2) : K = 128 ----
    {
        const int c = wave * 16 + row;
        v8f acc0 = {}, acc1 = {};
        const float2* Ar = (const float2*)(h1s + row * D_ATT);
        for (int k = 0; k < D_ATT; k += 8) {
            float2 av0 = Ar[(k >> 1) + hi];
            float2 av1 = Ar[(k >> 1) + 2 + hi];
            v2f a0, a1, bb0, bb1;
            a0.x = av0.x; a0.y = av0.y;
            a1.x = av1.x; a1.y = av1.y;
            bb0.x = W2[(size_t)(k + 2 * hi)     * D_ATT + c];
            bb0.y = W2[(size_t)(k + 2 * hi + 1) * D_ATT + c];
            bb1.x = W2[(size_t)(k + 4 + 2 * hi)     * D_ATT + c];
            bb1.y = W2[(size_t)(k + 4 + 2 * hi + 1) * D_ATT + c];
            acc0 = __builtin_amdgcn_wmma_f32_16x16x4_f32(false, a0, false, bb0,
                                                         (short)0, acc0, false, false);
            acc1 = __builtin_amdgcn_wmma_f32_16x16x4_f32(false, a1, false, bb1,
                                                         (short)0, acc1, false, false);
        }
        #pragma unroll
        for (int r = 0; r < 8; ++r) {
            float v = acc0[r] + acc1[r] + b2[c];
            h2s[(r + 8 * hi) * D_ATT + c] = v > 0.f ? v : 0.f;
        }
    }
    __syncthreads();

    // ---- params = h2 @ W3 : 24 cols -> waves 0,1 (wave-uniform branch) ----
    if (wave < 2) {
        const int c = wave * 16 + row;
        const bool valid = (c < 3 * MMIX);
        v8f acc = {};
        const float2* Ar = (const float2*)(h2s + row * D_ATT);
        for (int k = 0; k < D_ATT; k += 4) {
            float2 av = Ar[(k >> 1) + hi];
            v2f a, bb;
            a.x = av.x; a.y = av.y;
            bb.x = valid ? W3[(size_t)(k + 2 * hi)     * (3 * MMIX) + c] : 0.f;
            bb.y = valid ? W3[(size_t)(k + 2 * hi + 1) * (3 * MMIX) + c] : 0.f;
            acc = __builtin_amdgcn_wmma_f32_16x16x4_f32(false, a, false, bb,
                                                        (short)0, acc, false, false);
        }
        if (valid) {
            #pragma unroll
            for (int r = 0; r < 8; ++r)
                prs[(r + 8 * hi) * 24 + c] = acc[r];
        }
    }
    __syncthreads();

    // ---- per-row postprocess: sigmoid / softmax / loc update ----
    if (tid < 16) {
        const int bi = rbase + tid;
        float w[MMIX];
        float mx = -3.0e38f;
        #pragma unroll
        for (int m = 0; m < MMIX; ++m) {
            w[m] = prs[tid * 24 + m];
            mx = fmaxf(mx, w[m]);
        }
        float se = 0.f;
        #pragma unroll
        for (int m = 0; m < MMIX; ++m) { w[m] = expf(w[m] - mx); se += w[m]; }
        const float inv_se = 1.f / se;
        #pragma unroll
        for (int m = 0; m < MMIX; ++m) {
            const float dl = prs[tid * 24 + MMIX + m];
            const float sc = prs[tid * 24 + 2 * MMIX + m];
            const float loc = prev_loc[bi * MMIX + m] + 1.f / (1.f + expf(-dl));
            const float stdv = 1.f / (1.f + expf(-sc)) * 5.f + 0.2f;
            out_loc[bi * MMIX + m]    = loc;
            ws_loc[bi * MMIX + m]     = loc;
            ws_inv_std[bi * MMIX + m] = 1.f / stdv;
            ws_wm[bi * MMIX + m]      = w[m] * inv_se;
        }
    }
}

// ---------------------------------------------------------------------------
// Kernel 2: alignment (erf CDF diff, masked) + streaming context partials.
// Grid: (4 T-chunks, 128 batches) x 256 threads. 512 blocks keep enough
// loads in flight to saturate HBM (the 512 MB `memory` read dominates).
// ---------------------------------------------------------------------------
__launch_bounds__(256)
__global__ void align_ctx_kernel(const float* __restrict__ memory,
                                 const unsigned char* __restrict__ mask,
                                 const float* __restrict__ ws_loc,
                                 const float* __restrict__ ws_inv_std,
                                 const float* __restrict__ ws_wm,
                                 float* __restrict__ out_align,
                                 float* __restrict__ partial)
{
    const int chunk = blockIdx.x;           // 0..3, 500 timesteps each
    const int b     = blockIdx.y;           // 0..127
    const int tid   = threadIdx.x;          // 0..255

    __shared__ float s_align[256];
    __shared__ float s_loc[MMIX], s_is[MMIX], s_wm[MMIX];
    if (tid < MMIX) {
        s_loc[tid] = ws_loc[b * MMIX + tid];
        s_is[tid]  = ws_inv_std[b * MMIX + tid];
        s_wm[tid]  = ws_wm[b * MMIX + tid];
    }
    __syncthreads();

    const int tstart = chunk * (T_LEN / 4);
    const int tend   = tstart + (T_LEN / 4);
    const float2* memb = (const float2*)(memory + (size_t)b * T_LEN * D_EMB);

    float2 acc0 = {0.f, 0.f}, acc1 = {0.f, 0.f};

    for (int t0 = tstart; t0 < tend; t0 += 256) {
        const int t = t0 + tid;
        float a = 0.f;
        if (t < tend) {
            const float pos = (float)t;
            #pragma unroll
            for (int m = 0; m < MMIX; ++m) {
                const float dd = s_loc[m] - pos;
                const float is = s_is[m];
                a += s_wm[m] * (erff((dd + 0.5f) * is) - erff((dd - 0.5f) * is));
            }
            a *= 0.5f;
            if (mask[(size_t)b * T_LEN + t]) a = 0.f;
            out_align[(size_t)b * T_LEN + t] = a;
        }
        __syncthreads();                 // previous phase-B reads done
        s_align[tid] = (t < tend) ? a : 0.f;
        __syncthreads();                 // broadcast visible

        const int tmax = (tend - t0) < 256 ? (tend - t0) : 256;
        int tt = 0;
        for (; tt + 4 <= tmax; tt += 4) {
            const float2* r0 = memb + (size_t)(t0 + tt) * (D_EMB / 2);
            if (t0 + tt + 16 < tend)     // stay in-bounds; speculative anyway
                __builtin_prefetch((const void*)((const float*)(r0 + 16 * (D_EMB / 2)) + 2 * tid), 0, 0);
            const float a0 = s_align[tt + 0];
            const float a1 = s_align[tt + 1];
            const float a2 = s_align[tt + 2];
            const float a3 = s_align[tt + 3];
            const float2 v0 = r0[tid];
            const float2 v1 = (r0 + 1 * (D_EMB / 2))[tid];
            const float2 v2 = (r0 + 2 * (D_EMB / 2))[tid];
            const float2 v3 = (r0 + 3 * (D_EMB / 2))[tid];
            acc0.x = fmaf(a0, v0.x, acc0.x); acc0.y = fmaf(a0, v0.y, acc0.y);
            acc1.x = fmaf(a1, v1.x, acc1.x); acc1.y = fmaf(a1, v1.y, acc1.y);
            acc0.x = fmaf(a2, v2.x, acc0.x); acc0.y = fmaf(a2, v2.y, acc0.y);
            acc1.x = fmaf(a3, v3.x, acc1.x); acc1.y = fmaf(a3, v3.y, acc1.y);
        }
        for (; tt < tmax; ++tt) {
            const float av = s_align[tt];
            const float2 v = (memb + (size_t)(t0 + tt) * (D_EMB / 2))[tid];
            acc0.x = fmaf(av, v.x, acc0.x); acc0.y = fmaf(av, v.y, acc0.y);
        }
    }

    float* pc = partial + (size_t)(b * 4 + chunk) * D_EMB;
    pc[2 * tid + 0] = acc0.x + acc1.x;
    pc[2 * tid + 1] = acc0.y + acc1.y;
}

// ---------------------------------------------------------------------------
// Kernel 3: reduce the 4 T-chunk partials into context (deterministic).
// ---------------------------------------------------------------------------
__launch_bounds__(256)
__global__ void reduce_ctx_kernel(const float* __restrict__ partial,
                                  float* __restrict__ out_ctx)
{
    const int i = blockIdx.x * 256 + threadIdx.x;   // 0 .. 128*512-1
    const int b = i >> 9;
    const int d = i & (D_EMB - 1);
    const float* p = partial + (size_t)b * 4 * D_EMB + d;
    out_ctx[i] = (p[0] + p[D_EMB]) + (p[2 * D_EMB] + p[3 * D_EMB]);
}

extern "C" void kernel_launch(void* const* d_in, const int* in_sizes, int n_in,
                              void* d_out, int out_size, void* d_ws, size_t ws_size,
                              hipStream_t stream) {
    const float*         X      = (const float*)d_in[0];
    const float*         memory = (const float*)d_in[1];
    const float*         prev   = (const float*)d_in[2];
    const unsigned char* mask   = (const unsigned char*)d_in[3];   // jnp bool = 1 byte
    const float*         W1     = (const float*)d_in[4];
    const float*         b1     = (const float*)d_in[5];
    const float*         W2     = (const float*)d_in[6];
    const float*         b2     = (const float*)d_in[7];
    const float*         W3     = (const float*)d_in[8];

    float* out       = (float*)d_out;
    float* out_ctx   = out;                                // 128*512
    float* out_align = out + NB * D_EMB;                   // 128*2000
    float* out_loc   = out + NB * D_EMB + NB * T_LEN;      // 128*8

    float* ws         = (float*)d_ws;
    float* ws_loc     = ws;                  // 1024 floats
    float* ws_inv_std = ws + NB * MMIX;      // 1024 floats
    float* ws_wm      = ws + 2 * NB * MMIX;  // 1024 floats
    float* ws_partial = ws + 3 * NB * MMIX;  // 128*4*512 floats (1 MB)

    mlp_wmma_kernel<<<NB / 16, 256, 0, stream>>>(X, prev, W1, b1, W2, b2, W3,
                                                 out_loc, ws_loc, ws_inv_std, ws_wm);

    dim3 g2(4, NB);
    align_ctx_kernel<<<g2, 256, 0, stream>>>(memory, mask, ws_loc, ws_inv_std, ws_wm,
                                             out_align, ws_partial);

    reduce_ctx_kernel<<<(NB * D_EMB) / 256, 256, 0, stream>>>(ws_partial, out_ctx);
}